// LBPLayer_2894807957921
// MI455X (gfx1250) — compile-verified
//
#include <hip/hip_runtime.h>
#include <cstdint>

// LBP layer for (32, 512, 512, 3) fp32 -> (32, 512, 512, 1) fp32.
// Memory-bound stencil (AI ~2.3 flop/byte; ~134MB @ 23.3TB/s => ~5.8us floor).
// Strategy: 32x32 output tile per workgroup (256 threads = 8 wave32).
//   Phase 1: async-copy raw RGB halo rows to LDS via CDNA5
//            global_load_async_to_lds_b128 (ASYNCcnt path), 16B-aligned chunks.
//   Phase 2: grayscale 36x36 into LDS (zero outside image).
//   Phase 3: 3x3 census sum s on 34x34 into LDS (zero outside image).
//   Phase 4: 3x3 weighted LBP stencil, 4 px/thread, float4 NT store.

#define TILE 32
#define GDIM 36            // TILE + 4 (2-pixel halo for gray)
#define SDIM 34            // TILE + 2 (1-pixel halo for census)
#define GSTR 37            // padded LDS stride (bank-conflict avoidance)
#define SSTR 35
#define NTHREADS 256
#define IMG 512

// Staging geometry: rows start at pixel x0-4 (multiple of 4 => byte offset
// px*12 is a multiple of 48 => 16B aligned). 40 pixels = 120 dwords =
// 30 x b128 chunks per row; gray halo col c maps to staged pixel (c+2).
#define LPX   40           // staged pixels per row
#define LDW   120          // staged dwords per row
#define LCH   30           // b128 chunks per row
#define NROWCH_SLOTS 32    // power-of-2 chunk slots per row (clamped to 29)

typedef float v4f __attribute__((ext_vector_type(4)));

__global__ __launch_bounds__(NTHREADS) void lbp_kernel(const float* __restrict__ in,
                                                       float* __restrict__ out)
{
    __shared__ __align__(16) float lds_rgb[GDIM * LDW];   // 17,280 B
    __shared__ float lds_gray[GDIM * GSTR];               //  5,328 B
    __shared__ float lds_s[SDIM * SSTR];                  //  4,760 B

    const int tid = threadIdx.x;
    const int x0  = blockIdx.x * TILE;
    const int y0  = blockIdx.y * TILE;
    const int n   = blockIdx.z;

    // ---------- Phase 1: async b128 copy of RGB halo rows ----------
    const int TOT_SLOTS    = GDIM * NROWCH_SLOTS;         // 1152
    const int IN_MAX_CHUNK = 32 * IMG * IMG * 3 / 4 - 1;  // 6,291,455
    const uint32_t lds_rgb_base = (uint32_t)(uintptr_t)&lds_rgb[0];

    #pragma unroll
    for (int i = 0; i < 5; ++i) {                         // 5*256 >= 1152
        int t = tid + i * NTHREADS;
        if (t > TOT_SLOTS - 1) t = TOT_SLOTS - 1;         // keep EXEC fully on
        int r  = t >> 5;                                  // halo row 0..35
        int ch = t & (NROWCH_SLOTS - 1);
        if (ch > LCH - 1) ch = LCH - 1;                   // duplicate tail chunk
        int gy = y0 - 2 + r;
        gy = gy < 0 ? 0 : (gy > IMG - 1 ? IMG - 1 : gy);
        // 16B-chunk index into the input buffer (exact integer arithmetic)
        int chunk = (n * IMG + gy) * (IMG * 12 / 16) + ((x0 - 4) * 3) / 4 + ch;
        chunk = chunk < 0 ? 0 : (chunk > IN_MAX_CHUNK ? IN_MAX_CHUNK : chunk);
        const float* gptr = in + (size_t)chunk * 4;       // 16B aligned
        uint32_t laddr = lds_rgb_base + (uint32_t)(r * LCH + ch) * 16u;
        // CDNA5 async global->LDS copy (128b), tracked by ASYNCcnt
        asm volatile("global_load_async_to_lds_b128 %0, %1, off"
                     :: "v"(laddr), "v"(gptr)
                     : "memory");
    }
    asm volatile("s_wait_asynccnt 0" ::: "memory");       // our async writes landed
    __syncthreads();                                       // visible to all 8 waves

    // ---------- Phase 2: grayscale (36x36), zero-padded ----------
    for (int t = tid; t < GDIM * GDIM; t += NTHREADS) {
        int r = t / GDIM;
        int c = t - r * GDIM;
        int gy = y0 - 2 + r, gx = x0 - 2 + c;
        float g = 0.0f;
        if ((unsigned)gy < (unsigned)IMG && (unsigned)gx < (unsigned)IMG) {
            int base = r * LDW + (c + 2) * 3;             // staged pixel (c+2)
            float R = lds_rgb[base + 0];
            float G = lds_rgb[base + 1];
            float B = lds_rgb[base + 2];
            g = 0.2989f * R + 0.587f * G + 0.114f * B;
        }
        lds_gray[r * GSTR + c] = g;
    }
    __syncthreads();

    // ---------- Phase 3: census sum s (34x34), zero outside image ----------
    for (int t = tid; t < SDIM * SDIM; t += NTHREADS) {
        int r = t / SDIM;
        int c = t - r * SDIM;
        int gy = y0 - 1 + r, gx = x0 - 1 + c;
        float s = 0.0f;
        if ((unsigned)gy < (unsigned)IMG && (unsigned)gx < (unsigned)IMG) {
            float ctr = lds_gray[(r + 1) * GSTR + (c + 1)];
            #pragma unroll
            for (int dy = 0; dy < 3; ++dy)
                #pragma unroll
                for (int dx = 0; dx < 3; ++dx)
                    s += (lds_gray[(r + dy) * GSTR + (c + dx)] >= ctr) ? 1.0f : 0.0f;
        }
        lds_s[r * SSTR + c] = s;
    }
    __syncthreads();

    // ---------- Phase 4: weighted LBP stencil, 4 px/thread ----------
    {
        int r  = tid >> 3;            // 0..31
        int c0 = (tid & 7) << 2;      // 0,4,...,28
        const float* s0 = &lds_s[(r + 0) * SSTR + c0];
        const float* s1 = &lds_s[(r + 1) * SSTR + c0];
        const float* s2 = &lds_s[(r + 2) * SSTR + c0];
        float acc[4];
        #pragma unroll
        for (int j = 0; j < 4; ++j) {
            acc[j] =   1.0f * s0[j] +   2.0f * s0[j + 1] +  4.0f * s0[j + 2]
                   + 128.0f * s1[j]                      +  8.0f * s1[j + 2]
                   +  64.0f * s2[j] +  32.0f * s2[j + 1] + 16.0f * s2[j + 2];
        }
        v4f v = { acc[0], acc[1], acc[2], acc[3] };
        int off = (n * IMG + (y0 + r)) * IMG + (x0 + c0);
        __builtin_nontemporal_store(v, reinterpret_cast<v4f*>(out + off));
    }
}

extern "C" void kernel_launch(void* const* d_in, const int* in_sizes, int n_in,
                              void* d_out, int out_size, void* d_ws, size_t ws_size,
                              hipStream_t stream) {
    const float* in = (const float*)d_in[0];
    float* out      = (float*)d_out;
    dim3 grid(IMG / TILE, IMG / TILE, 32);   // 16 x 16 x 32 = 8192 workgroups
    dim3 block(NTHREADS);
    lbp_kernel<<<grid, block, 0, stream>>>(in, out);
}